// IterLocalSoftmaxAttention_32719060861269
// MI455X (gfx1250) — compile-verified
//
#include <hip/hip_runtime.h>

// IterLocalSoftmaxAttention for gfx1250 (MI455X).
// B=4, C_IN=C_HID=C_OUT=128, N=10000, K=32. All fp32.
// Pipeline:
//   1) qkv_proj   : (B*N,128)x(128,128) GEMMs via V_WMMA_F32_16X16X4_F32 -> q,k,v in d_ws
//   2) local_attn : per-query gather(32 neighbors) + scores + softmax + weighted V (wave32 VALU/shfl)
//   3) final_proj : attn @ Wf + bf via WMMA, transposed store to (B, C_OUT, N)

#define NPTS 10000
#define BATCH 4
#define CH 128
#define KNBR 32

typedef float v2f __attribute__((ext_vector_type(2)));
typedef float v8f __attribute__((ext_vector_type(8)));

__device__ __forceinline__ v8f wmma_f32_4(v2f a, v2f b, v8f c) {
  // D(16x16,f32) = A(16x4,f32) x B(4x16,f32) + C
  return __builtin_amdgcn_wmma_f32_16x16x4_f32(
      /*neg_a=*/false, a, /*neg_b=*/false, b,
      /*c_mod=*/(short)0, c, /*reuse_a=*/false, /*reuse_b=*/false);
}

// ---------------------------------------------------------------------------
// Kernel 1: q/k/v projections.
// grid = (rowTiles=2500, 3), block = 256 (8 waves; wave w covers out channels [16w,16w+16)).
// A layout (f32 16x4): VGPR p, lane (h*16+m) holds A[m][2h+p]; A[m][k] = x[b, c0+k, n0+m].
// B layout (4x16):     VGPR p, lane (h*16+m) holds B[2h+p][m] = W[(c0+2h+p)*128 + jBase+m].
// C/D layout:          VGPR i, lane (h*16+m) = D[i+8h][m].
// ---------------------------------------------------------------------------
__global__ void qkv_proj(const float* __restrict__ x,
                         const float* __restrict__ Wq, const float* __restrict__ bq,
                         const float* __restrict__ Wk, const float* __restrict__ bk,
                         const float* __restrict__ Wv, const float* __restrict__ bv,
                         float* __restrict__ qo, float* __restrict__ ko,
                         float* __restrict__ vo) {
  const float* W;
  const float* bias;
  float* out;
  if (blockIdx.y == 0)      { W = Wq; bias = bq; out = qo; }
  else if (blockIdx.y == 1) { W = Wk; bias = bk; out = ko; }
  else                      { W = Wv; bias = bv; out = vo; }

  const int lane  = threadIdx.x & 31;
  const int wid   = threadIdx.x >> 5;
  const int half  = lane >> 4;
  const int m     = lane & 15;
  const int jBase = wid * 16;

  const int r0 = blockIdx.x * 16;   // N % 16 == 0, tiles never straddle batches
  const int b  = r0 / NPTS;
  const int n0 = r0 - b * NPTS;
  const float* xb = x + (size_t)b * CH * NPTS;

  v8f acc = {};
#pragma unroll
  for (int c0 = 0; c0 < CH; c0 += 4) {
    const int ka = c0 + half * 2;
    v2f a, bm;
    a.x  = xb[(size_t)(ka + 0) * NPTS + n0 + m];
    a.y  = xb[(size_t)(ka + 1) * NPTS + n0 + m];
    bm.x = W[(size_t)(ka + 0) * CH + jBase + m];
    bm.y = W[(size_t)(ka + 1) * CH + jBase + m];
    acc = wmma_f32_4(a, bm, acc);
  }

  const float bb = bias[jBase + m];
#pragma unroll
  for (int i = 0; i < 8; ++i) {
    const int row = r0 + i + half * 8;
    out[(size_t)row * CH + jBase + m] = acc[i] + bb;
  }
}

// ---------------------------------------------------------------------------
// Kernel 2: gather + scores + softmax + weighted sum. One wave per (b, n).
// grid = 40000/8 blocks of 256 threads (8 waves).
// attn output aliases the q buffer (each wave reads/writes only its own row).
// ---------------------------------------------------------------------------
__global__ void local_attn(const float* qbuf, const float* __restrict__ kbuf,
                           const float* __restrict__ vbuf,
                           const int* __restrict__ nbr, float* attn) {
  const int lane = threadIdx.x & 31;
  const int wid  = threadIdx.x >> 5;
  const int r    = blockIdx.x * (blockDim.x >> 5) + wid;  // 0..B*N-1 (exact)
  const int b    = r / NPTS;
  const int n    = r - b * NPTS;

  // Lane l holds q dims [4l, 4l+4)
  const float4 q4 = ((const float4*)(qbuf + (size_t)r * CH))[lane];

  // Lane l owns neighbor l
  const int idx  = nbr[n * KNBR + lane];
  const int krow = b * NPTS + idx;

  // scores: s_l = q . k[krow_l]
  const float4* kp = (const float4*)(kbuf + (size_t)krow * CH);
  float s = 0.f;
#pragma unroll
  for (int j = 0; j < 32; ++j) {
    const float4 kv = kp[j];
    const float qx = __shfl(q4.x, j, 32);
    const float qy = __shfl(q4.y, j, 32);
    const float qz = __shfl(q4.z, j, 32);
    const float qw = __shfl(q4.w, j, 32);
    s += qx * kv.x + qy * kv.y + qz * kv.z + qw * kv.w;
  }
  s *= 0.088388347648318447f;  // 1/sqrt(128)

  // softmax over the 32 lanes
  float mx = s;
#pragma unroll
  for (int off = 16; off >= 1; off >>= 1)
    mx = fmaxf(mx, __shfl_xor(mx, off, 32));
  const float p = __expf(s - mx);
  float sum = p;
#pragma unroll
  for (int off = 16; off >= 1; off >>= 1)
    sum += __shfl_xor(sum, off, 32);
  const float w = p / sum;

  // attn_h = sum_k w_k * v[krow_k][h]; lane l covers dims [4l, 4l+4)
  float4 acc = make_float4(0.f, 0.f, 0.f, 0.f);
#pragma unroll
  for (int j = 0; j < 32; ++j) {
    const float wj = __shfl(w, j, 32);
    const int   kr = __shfl(krow, j, 32);
    const float4 vv = ((const float4*)(vbuf + (size_t)kr * CH))[lane];
    acc.x += wj * vv.x;
    acc.y += wj * vv.y;
    acc.z += wj * vv.z;
    acc.w += wj * vv.w;
  }
  ((float4*)(attn + (size_t)r * CH))[lane] = acc;
}

// ---------------------------------------------------------------------------
// Kernel 3: out[b, c, n] = (attn @ Wf + bf)[b*N+n, c]  (transposed store).
// grid = 2500 blocks of 256 (8 waves, wave w -> out channels [16w, 16w+16)).
// ---------------------------------------------------------------------------
__global__ void final_proj(const float* __restrict__ attn,
                           const float* __restrict__ Wf,
                           const float* __restrict__ bf,
                           float* __restrict__ out) {
  const int lane  = threadIdx.x & 31;
  const int wid   = threadIdx.x >> 5;
  const int half  = lane >> 4;
  const int m     = lane & 15;
  const int jBase = wid * 16;

  const int r0 = blockIdx.x * 16;
  const int b  = r0 / NPTS;
  const int n0 = r0 - b * NPTS;

  v8f acc = {};
#pragma unroll
  for (int c0 = 0; c0 < CH; c0 += 4) {
    const int ka = c0 + half * 2;
    v2f a, bm;
    a.x  = attn[(size_t)(r0 + m) * CH + ka + 0];
    a.y  = attn[(size_t)(r0 + m) * CH + ka + 1];
    bm.x = Wf[(size_t)(ka + 0) * CH + jBase + m];
    bm.y = Wf[(size_t)(ka + 1) * CH + jBase + m];
    acc = wmma_f32_4(a, bm, acc);
  }

  const float bb = bf[jBase + m];
  float* ob = out + (size_t)b * CH * NPTS + (size_t)(jBase + m) * NPTS;
#pragma unroll
  for (int i = 0; i < 8; ++i) {
    ob[n0 + i + half * 8] = acc[i] + bb;  // out[b, jBase+m, n0+M]
  }
}

extern "C" void kernel_launch(void* const* d_in, const int* in_sizes, int n_in,
                              void* d_out, int out_size, void* d_ws, size_t ws_size,
                              hipStream_t stream) {
  const float* x   = (const float*)d_in[0];
  const int*   nbr = (const int*)d_in[1];  // jax default x64-off: int32
  const float* Wq  = (const float*)d_in[2];
  const float* bq  = (const float*)d_in[3];
  const float* Wk  = (const float*)d_in[4];
  const float* bk  = (const float*)d_in[5];
  const float* Wv  = (const float*)d_in[6];
  const float* bv  = (const float*)d_in[7];
  const float* Wf  = (const float*)d_in[8];
  const float* bf  = (const float*)d_in[9];
  float* out = (float*)d_out;

  const size_t perBuf = (size_t)BATCH * NPTS * CH;  // 5.12M floats
  float* qb = (float*)d_ws;
  float* kb = qb + perBuf;
  float* vb = kb + perBuf;
  float* at = qb;  // attn reuses the q region (row-local read-then-write)

  const int rowTiles = (BATCH * NPTS) / 16;  // 2500

  qkv_proj<<<dim3(rowTiles, 3), 256, 0, stream>>>(x, Wq, bq, Wk, bk, Wv, bv,
                                                  qb, kb, vb);
  local_attn<<<(BATCH * NPTS) / 8, 256, 0, stream>>>(qb, kb, vb, nbr, at);
  final_proj<<<rowTiles, 256, 0, stream>>>(at, Wf, bf, out);
}